// VanillaGNN_53446573032076
// MI455X (gfx1250) — compile-verified
//
#include <hip/hip_runtime.h>
#include <hip/hip_bf16.h>
#include <cmath>

typedef __attribute__((ext_vector_type(2))) float v2f;
typedef __attribute__((ext_vector_type(8))) float v8f;

#define N_NODES 50000
#define N_EDGES 800000
#define DIM_IN  128
#define DIM_H   64
#define DIM_OUT 16

// ---------------------------------------------------------------------------
// GEMM via V_WMMA_F32_16X16X4_F32:  out[M x N] = act(A[M x K]) @ W[K x N]
// One wave computes one 16x16 f32 tile. MT m-tiles x (N/16) n-tiles per block.
// A-frag (16x4): lane<16 -> M=lane, {K=k,k+1}; lane>=16 -> M=lane-16, {K=k+2,k+3}
// B-frag (4x16): lane%16 = N col, same K split across lane halves.
// C/D: VGPR r -> M = r + 8*(lane>=16), N = lane%16.
// ---------------------------------------------------------------------------
template<int K, int N, int MT, bool TANH>
__global__ __launch_bounds__(32 * MT * (N / 16))
void gemm_wmma_kernel(const float* __restrict__ A, const float* __restrict__ W,
                      float* __restrict__ out, int M)
{
    constexpr int NT = N / 16;
    const int wave = threadIdx.x >> 5;
    const int lane = threadIdx.x & 31;
    const int l16  = lane & 15;
    const int hi   = lane >> 4;            // 0: lanes 0-15, 1: lanes 16-31
    const int mtile = wave / NT;
    const int ntile = wave % NT;
    const int row0  = (blockIdx.x * MT + mtile) * 16;
    const int col   = ntile * 16 + l16;

    int mA = row0 + l16;                   // clamp reads; stores are guarded
    if (mA > M - 1) mA = M - 1;
    const float* __restrict__ arow = A + (size_t)mA * K;

    v8f c = {0.f, 0.f, 0.f, 0.f, 0.f, 0.f, 0.f, 0.f};

#pragma unroll
    for (int kk = 0; kk < K; kk += 4) {
        const int k0 = kk + 2 * hi;        // even -> 8B-aligned v2f load
        v2f a = *(const v2f*)(arow + k0);
        if (TANH) { a.x = tanhf(a.x); a.y = tanhf(a.y); }
        v2f b;
        b.x = W[(size_t)k0 * N + col];
        b.y = W[(size_t)(k0 + 1) * N + col];
        c = __builtin_amdgcn_wmma_f32_16x16x4_f32(
                /*neg_a=*/false, a, /*neg_b=*/false, b,
                /*c_mod=*/(short)0, c, /*reuse_a=*/false, /*reuse_b=*/false);
    }

#pragma unroll
    for (int r = 0; r < 8; ++r) {
        const int m = row0 + r + 8 * hi;
        if (m < M) out[(size_t)m * N + col] = c[r];
    }
}

// ---------------------------------------------------------------------------
// SpMM scatter: out[dst] += val * h[src]  (COO).  16 (or 4) threads per edge,
// float4 gather (L2-resident) + 4 native global_atomic_add_f32 per thread.
// ---------------------------------------------------------------------------
template<int DIM>
__global__ __launch_bounds__(256)
void spmm_kernel(const int* __restrict__ esrc, const int* __restrict__ edst,
                 const float* __restrict__ evals, const float* __restrict__ h,
                 float* __restrict__ out)
{
    constexpr int CH = DIM / 4;
    const long long t = (long long)blockIdx.x * blockDim.x + threadIdx.x;
    const int e = (int)(t / CH);
    const int c = (int)(t % CH);
    if (e >= N_EDGES) return;
    const int   s = esrc[e];
    const int   d = edst[e];
    const float v = evals[e];
    const float4 hv = *(const float4*)(h + (size_t)s * DIM + c * 4);
    float* o = out + (size_t)d * DIM + c * 4;
    unsafeAtomicAdd(o + 0, hv.x * v);
    unsafeAtomicAdd(o + 1, hv.y * v);
    unsafeAtomicAdd(o + 2, hv.z * v);
    unsafeAtomicAdd(o + 3, hv.w * v);
}

// ---------------------------------------------------------------------------
// Row softmax over DIM_OUT=16 columns; one thread per row.
// ---------------------------------------------------------------------------
__global__ __launch_bounds__(256)
void softmax16_kernel(const float* __restrict__ in, float* __restrict__ out)
{
    const int r = blockIdx.x * blockDim.x + threadIdx.x;
    if (r >= N_NODES) return;
    const float* row = in + (size_t)r * DIM_OUT;
    float v[DIM_OUT];
    float mx = -3.402823466e38f;
#pragma unroll
    for (int i = 0; i < DIM_OUT; ++i) { v[i] = row[i]; mx = fmaxf(mx, v[i]); }
    float s = 0.f;
#pragma unroll
    for (int i = 0; i < DIM_OUT; ++i) { v[i] = __expf(v[i] - mx); s += v[i]; }
    const float inv = 1.f / s;
    float* orow = out + (size_t)r * DIM_OUT;
#pragma unroll
    for (int i = 0; i < DIM_OUT; ++i) orow[i] = v[i] * inv;
}

// ---------------------------------------------------------------------------
// Orchestration. Workspace layout (needs 2 * 12.8 MB = 25.6 MB):
//   bufA: h1 / h2 / h3   bufB: g1 / g2 / g3 (accumulators, memset before use)
// ---------------------------------------------------------------------------
extern "C" void kernel_launch(void* const* d_in, const int* in_sizes, int n_in,
                              void* d_out, int out_size, void* d_ws, size_t ws_size,
                              hipStream_t stream)
{
    const float* x    = (const float*)d_in[0];
    const int*   esrc = (const int*)  d_in[1];
    const int*   edst = (const int*)  d_in[2];
    const float* ev   = (const float*)d_in[3];
    const float* W1   = (const float*)d_in[4];
    const float* W2   = (const float*)d_in[5];
    const float* W3   = (const float*)d_in[6];
    float* outp = (float*)d_out;

    float* bufA = (float*)d_ws;
    float* bufB = bufA + (size_t)N_NODES * DIM_H;

    // Layer 1: h1 = x @ W1  -> bufA
    {
        dim3 grid((N_NODES + 31) / 32);
        gemm_wmma_kernel<DIM_IN, DIM_H, 2, false>
            <<<grid, 256, 0, stream>>>(x, W1, bufA, N_NODES);
    }
    hipMemsetAsync(bufB, 0, (size_t)N_NODES * DIM_H * sizeof(float), stream);
    {   // g1 = A_sp @ h1 -> bufB
        long long tt = (long long)N_EDGES * (DIM_H / 4);
        dim3 grid((unsigned)((tt + 255) / 256));
        spmm_kernel<DIM_H><<<grid, 256, 0, stream>>>(esrc, edst, ev, bufA, bufB);
    }

    // Layer 2: h2 = tanh(g1) @ W2 -> bufA
    {
        dim3 grid((N_NODES + 31) / 32);
        gemm_wmma_kernel<DIM_H, DIM_H, 2, true>
            <<<grid, 256, 0, stream>>>(bufB, W2, bufA, N_NODES);
    }
    hipMemsetAsync(bufB, 0, (size_t)N_NODES * DIM_H * sizeof(float), stream);
    {   // g2 = A_sp @ h2 -> bufB
        long long tt = (long long)N_EDGES * (DIM_H / 4);
        dim3 grid((unsigned)((tt + 255) / 256));
        spmm_kernel<DIM_H><<<grid, 256, 0, stream>>>(esrc, edst, ev, bufA, bufB);
    }

    // Layer 3: h3 = g2 @ W3 -> bufA (16-wide)
    {
        dim3 grid((N_NODES + 127) / 128);
        gemm_wmma_kernel<DIM_H, DIM_OUT, 8, false>
            <<<grid, 256, 0, stream>>>(bufB, W3, bufA, N_NODES);
    }
    hipMemsetAsync(bufB, 0, (size_t)N_NODES * DIM_OUT * sizeof(float), stream);
    {   // g3 = A_sp @ h3 -> bufB
        long long tt = (long long)N_EDGES * (DIM_OUT / 4);
        dim3 grid((unsigned)((tt + 255) / 256));
        spmm_kernel<DIM_OUT><<<grid, 256, 0, stream>>>(esrc, edst, ev, bufA, bufB);
    }

    // softmax(g3) -> d_out
    {
        dim3 grid((N_NODES + 255) / 256);
        softmax16_kernel<<<grid, 256, 0, stream>>>(bufB, outp);
    }
}